// RGCNNet_57767310131499
// MI455X (gfx1250) — compile-verified
//
#include <hip/hip_runtime.h>

// Problem constants (from reference)
#define NN   50000
#define EE   800000
#define RR   50
#define FIN  768
#define HID  256
#define CC   9
#define KEEP 0.6f

typedef __attribute__((ext_vector_type(2))) float v2f;
typedef __attribute__((ext_vector_type(8))) float v8f;

// ---------------------------------------------------------------------------
// JAX threefry2x32 (exact port of jax._src.prng.threefry2x32)
// ---------------------------------------------------------------------------
__host__ __device__ inline unsigned rotl32(unsigned x, unsigned r) {
  return (x << r) | (x >> (32u - r));
}

__host__ __device__ inline void threefry2x32(unsigned k0, unsigned k1,
                                             unsigned x0, unsigned x1,
                                             unsigned* y0, unsigned* y1) {
  const unsigned ks2 = k0 ^ k1 ^ 0x1BD11BDAu;
  const unsigned ks[3] = {k0, k1, ks2};
  const unsigned rot[2][4] = {{13u, 15u, 26u, 6u}, {17u, 29u, 16u, 24u}};
  x0 += k0; x1 += k1;
#pragma unroll
  for (int i = 0; i < 5; ++i) {
    const unsigned* r = rot[i & 1];
#pragma unroll
    for (int j = 0; j < 4; ++j) {
      x0 += x1; x1 = rotl32(x1, r[j]); x1 ^= x0;
    }
    x0 += ks[(i + 1) % 3];
    x1 += ks[(i + 2) % 3] + (unsigned)(i + 1);
  }
  *y0 = x0; *y1 = x1;
}

// jax.random.uniform element i of a flat array of n (even) elements:
// counts = iota(n), split at n/2, element i pairs with (i%half, half+i%half).
__device__ inline float jax_uniform_elem(unsigned k0, unsigned k1,
                                         unsigned i, unsigned halfn) {
  unsigned j = (i < halfn) ? i : (i - halfn);
  unsigned y0, y1;
  threefry2x32(k0, k1, j, halfn + j, &y0, &y1);
  unsigned bits = (i < halfn) ? y0 : y1;
  return __uint_as_float((bits >> 9) | 0x3f800000u) - 1.0f;
}

// ---------------------------------------------------------------------------
// WMMA helper: full-precision fp32 matrix op (V_WMMA_F32_16X16X4_F32)
// A 16x4: lane half selects K{0,1} vs K{2,3}; B 4x16: lane = N, VGPR/half = K.
// ---------------------------------------------------------------------------
__device__ inline v8f wmma_f32(v2f a, v2f b, v8f c) {
  return __builtin_amdgcn_wmma_f32_16x16x4_f32(false, a, false, b,
                                               (short)0, c, false, false);
}

// ---------------------------------------------------------------------------
// Kernel 1: per-(dst,relation) edge counts
// ---------------------------------------------------------------------------
__global__ __launch_bounds__(256) void rgcn_count_kernel(
    const int* __restrict__ dst, const int* __restrict__ etype,
    unsigned* __restrict__ cnt) {
  int e = blockIdx.x * blockDim.x + threadIdx.x;
  if (e >= EE) return;
  atomicAdd(&cnt[(size_t)dst[e] * RR + etype[e]], 1u);
}

// ---------------------------------------------------------------------------
// Kernel 2: GEMM1 — hx = x @ basis1,  acc = x @ root1 + bias1   (K = 768)
// Block = 8 waves on one 16-row tile; waves 0..3 -> basis cols, 4..7 -> root.
// Each wave: 16x64 strip, 4 v8f accumulators, fp32 WMMA, direct global loads
// (x and weights are L2-resident on MI455X's 192MB L2).
// ---------------------------------------------------------------------------
__global__ __launch_bounds__(256) void rgcn_gemm1_kernel(
    const float* __restrict__ x,      // [N, 768]
    const float* __restrict__ basis,  // [768, 256]
    const float* __restrict__ root,   // [768, 256]
    const float* __restrict__ bias,   // [256]
    float* __restrict__ hx,           // [N, 256]
    float* __restrict__ acc) {        // [N, 256]
  const int lane = threadIdx.x & 31;
  const int wave = threadIdx.x >> 5;       // 0..7
  const int m0   = blockIdx.x * 16;        // N = 50000 = 3125 * 16 exactly
  const bool isRoot = (wave >= 4);
  const float* __restrict__ W = isRoot ? root : basis;
  float* __restrict__ OUT = isRoot ? acc : hx;
  const int col0 = (wave & 3) * 64;
  const int mr   = lane & 15;
  const int half = lane >> 4;              // 0/1 -> K pair {0,1} vs {2,3}

  v8f c0 = {}, c1 = {}, c2 = {}, c3 = {};
  const float* __restrict__ arow = x + (size_t)(m0 + mr) * FIN + half * 2;

#pragma unroll 4
  for (int k = 0; k < FIN; k += 4) {
    v2f a;
    a.x = arow[k];
    a.y = arow[k + 1];
    const float* __restrict__ w0 = W + (size_t)(k + half * 2) * HID + col0 + mr;
    const float* __restrict__ w1 = w0 + HID;
    v2f b;
    b.x = w0[0];  b.y = w1[0];  c0 = wmma_f32(a, b, c0);
    b.x = w0[16]; b.y = w1[16]; c1 = wmma_f32(a, b, c1);
    b.x = w0[32]; b.y = w1[32]; c2 = wmma_f32(a, b, c2);
    b.x = w0[48]; b.y = w1[48]; c3 = wmma_f32(a, b, c3);
  }

  // D layout: VGPR j holds row (j + half*8), lane%16 = column.
#pragma unroll
  for (int t = 0; t < 4; ++t) {
    const v8f cc = (t == 0) ? c0 : (t == 1) ? c1 : (t == 2) ? c2 : c3;
    const int ncol = col0 + t * 16 + mr;
    const float bv = isRoot ? bias[ncol] : 0.0f;
#pragma unroll
    for (int j = 0; j < 8; ++j)
      OUT[(size_t)(m0 + half * 8 + j) * HID + ncol] = cc[j] + bv;
  }
}

// ---------------------------------------------------------------------------
// Kernel 3: edge scatter layer 1 — one block (256 thr) per edge
// acc[dst] += comp[type] / cnt[dst,type] * hx[src]
// ---------------------------------------------------------------------------
__global__ __launch_bounds__(256) void rgcn_scatter1_kernel(
    const int* __restrict__ src, const int* __restrict__ dst,
    const int* __restrict__ etype, const unsigned* __restrict__ cnt,
    const float* __restrict__ comp,   // [R,1]
    const float* __restrict__ hx,     // [N, 256]
    float* __restrict__ acc) {        // [N, 256]
  const int e = blockIdx.x;
  const int s = src[e], d = dst[e], t = etype[e];
  const float norm  = 1.0f / fmaxf((float)cnt[(size_t)d * RR + t], 1.0f);
  const float coeff = comp[t] * norm;
  const int c = threadIdx.x;
  atomicAdd(&acc[(size_t)d * HID + c], coeff * hx[(size_t)s * HID + c]);
}

// ---------------------------------------------------------------------------
// Kernel 4: dropout(key1) + relu over [N,256] -> h1
// ---------------------------------------------------------------------------
__global__ __launch_bounds__(256) void rgcn_post1_kernel(
    const float* __restrict__ acc, float* __restrict__ h,
    unsigned k0, unsigned k1) {
  const unsigned n = NN * HID;
  unsigned i = blockIdx.x * blockDim.x + threadIdx.x;
  if (i >= n) return;
  float u = jax_uniform_elem(k0, k1, i, n / 2u);
  float v = acc[i];
  v = (u < KEEP) ? v * (1.0f / KEEP) : 0.0f;
  h[i] = fmaxf(v, 0.0f);
}

// ---------------------------------------------------------------------------
// Kernel 5: GEMM2 — hx2 = h1 @ basis2, acc2 = h1 @ root2 + bias2  (K=256, C=9)
// Output padded to width 16; lanes n>=9 masked via a 0/1 multiplier so EXEC
// stays all-ones at every WMMA.
// ---------------------------------------------------------------------------
__global__ __launch_bounds__(256) void rgcn_gemm2_kernel(
    const float* __restrict__ h1,     // [N, 256]
    const float* __restrict__ basis,  // [256, 9]
    const float* __restrict__ root,   // [256, 9]
    const float* __restrict__ bias,   // [9]
    float* __restrict__ hx2,          // [N, 16]
    float* __restrict__ acc2) {       // [N, 16]
  const int lane = threadIdx.x & 31;
  const int wave = threadIdx.x >> 5;
  const int gw = blockIdx.x * 8 + wave;
  const int mtile = gw >> 1;
  if (mtile >= NN / 16) return;                 // uniform per-wave exit
  const bool isRoot = (gw & 1);
  const float* __restrict__ W = isRoot ? root : basis;
  float* __restrict__ OUT = isRoot ? acc2 : hx2;
  const int m0 = mtile * 16;
  const int nl = lane & 15;
  const int half = lane >> 4;
  const int nc = (nl < CC) ? nl : (CC - 1);     // clamped, in-bounds
  const float mask = (nl < CC) ? 1.0f : 0.0f;

  v8f c = {};
  const float* __restrict__ arow = h1 + (size_t)(m0 + nl) * HID + half * 2;
#pragma unroll 8
  for (int k = 0; k < HID; k += 4) {
    v2f a;
    a.x = arow[k];
    a.y = arow[k + 1];
    v2f b;
    b.x = W[(size_t)(k + half * 2) * CC + nc] * mask;
    b.y = W[(size_t)(k + half * 2 + 1) * CC + nc] * mask;
    c = wmma_f32(a, b, c);
  }
  if (nl < CC) {
    const float bv = isRoot ? bias[nl] : 0.0f;
#pragma unroll
    for (int j = 0; j < 8; ++j)
      OUT[(size_t)(m0 + half * 8 + j) * 16 + nl] = c[j] + bv;
  }
}

// ---------------------------------------------------------------------------
// Kernel 6: edge scatter layer 2 — thread per (edge, col<9)
// ---------------------------------------------------------------------------
__global__ __launch_bounds__(256) void rgcn_scatter2_kernel(
    const int* __restrict__ src, const int* __restrict__ dst,
    const int* __restrict__ etype, const unsigned* __restrict__ cnt,
    const float* __restrict__ comp,   // [R,1]
    const float* __restrict__ hx2,    // [N, 16]
    float* __restrict__ acc2) {       // [N, 16]
  const long long tid = (long long)blockIdx.x * blockDim.x + threadIdx.x;
  const int e = (int)(tid >> 4);
  const int c = (int)(tid & 15);
  if (e >= EE || c >= CC) return;
  const int s = src[e], d = dst[e], t = etype[e];
  const float norm  = 1.0f / fmaxf((float)cnt[(size_t)d * RR + t], 1.0f);
  const float coeff = comp[t] * norm;
  atomicAdd(&acc2[(size_t)d * 16 + c], coeff * hx2[(size_t)s * 16 + c]);
}

// ---------------------------------------------------------------------------
// Kernel 7: dropout(key2) + relu over [N,9] -> d_out
// ---------------------------------------------------------------------------
__global__ __launch_bounds__(256) void rgcn_post2_kernel(
    const float* __restrict__ acc2, float* __restrict__ out,
    unsigned k0, unsigned k1) {
  const unsigned n = NN * CC;
  unsigned i = blockIdx.x * blockDim.x + threadIdx.x;
  if (i >= n) return;
  const unsigned row = i / CC;
  const unsigned col = i - row * CC;
  float u = jax_uniform_elem(k0, k1, i, n / 2u);
  float v = acc2[(size_t)row * 16 + col];
  v = (u < KEEP) ? v * (1.0f / KEEP) : 0.0f;
  out[i] = fmaxf(v, 0.0f);
}

// ---------------------------------------------------------------------------
extern "C" void kernel_launch(void* const* d_in, const int* in_sizes, int n_in,
                              void* d_out, int out_size, void* d_ws, size_t ws_size,
                              hipStream_t stream) {
  (void)in_sizes; (void)n_in; (void)out_size; (void)ws_size;

  const float* x       = (const float*)d_in[0];
  const int*   eidx    = (const int*)d_in[1];     // [2, E]
  const int*   etype   = (const int*)d_in[2];
  // d_in[3] edge_distance: unused by the reference
  const float* basis1  = (const float*)d_in[4];   // [1,768,256] -> [768,256]
  const float* comp1   = (const float*)d_in[5];   // [R,1]
  const float* root1   = (const float*)d_in[6];   // [768,256]
  const float* bias1   = (const float*)d_in[7];   // [256]
  const float* basis2  = (const float*)d_in[8];   // [1,256,9] -> [256,9]
  const float* comp2   = (const float*)d_in[9];   // [R,1]
  const float* root2   = (const float*)d_in[10];  // [256,9]
  const float* bias2   = (const float*)d_in[11];  // [9]
  float* out = (float*)d_out;

  const int* src = eidx;
  const int* dst = eidx + EE;

  // ---- workspace layout (bytes) ----
  char* ws = (char*)d_ws;
  unsigned* cnt = (unsigned*)ws;                              // 10,000,000
  float* hx1  = (float*)(ws + 10000000);                      // 51,200,000
  float* acc1 = (float*)(ws + 10000000 + 51200000);           // 51,200,000
  float* hx2  = (float*)(ws + 10000000 + 2 * 51200000);       //  3,200,000
  float* acc2 = (float*)(ws + 10000000 + 2 * 51200000 + 3200000);

  // ---- JAX key derivation on host: split(key(42)) -> k1, k2 ----
  unsigned a0, a1, b0, b1;
  threefry2x32(0u, 42u, 0u, 2u, &a0, &a1);
  threefry2x32(0u, 42u, 1u, 3u, &b0, &b1);
  const unsigned k1_hi = a0, k1_lo = b0;   // keys[0]
  const unsigned k2_hi = a1, k2_lo = b1;   // keys[1]

  hipMemsetAsync(cnt, 0, (size_t)NN * RR * sizeof(unsigned), stream);

  rgcn_count_kernel<<<(EE + 255) / 256, 256, 0, stream>>>(dst, etype, cnt);

  rgcn_gemm1_kernel<<<NN / 16, 256, 0, stream>>>(x, basis1, root1, bias1,
                                                 hx1, acc1);

  rgcn_scatter1_kernel<<<EE, 256, 0, stream>>>(src, dst, etype, cnt, comp1,
                                               hx1, acc1);

  rgcn_post1_kernel<<<(NN * HID + 255) / 256, 256, 0, stream>>>(
      acc1, hx1, k1_hi, k1_lo);  // h1 overwrites hx1

  rgcn_gemm2_kernel<<<((NN / 16) * 2 + 7) / 8, 256, 0, stream>>>(
      hx1, basis2, root2, bias2, hx2, acc2);

  rgcn_scatter2_kernel<<<((long long)EE * 16 + 255) / 256, 256, 0, stream>>>(
      src, dst, etype, cnt, comp2, hx2, acc2);

  rgcn_post2_kernel<<<(NN * CC + 255) / 256, 256, 0, stream>>>(
      acc2, out, k2_hi, k2_lo);
}